// HetSANNConv_22479858827461
// MI455X (gfx1250) — compile-verified
//
#include <hip/hip_runtime.h>
#include <hip/hip_bf16.h>
#include <math.h>

// ---- problem constants (match reference) ----
#define NTYPES   5
#define NHEADS   4
#define HID      32
#define INDIM    128
#define OUTDIM   128      // NHEADS*HID
#define NEGSLOPE 0.2f

typedef __attribute__((ext_vector_type(2))) float v2f;
typedef __attribute__((ext_vector_type(8))) float v8f;

// monotonic unsigned encoding for float atomic-max
__device__ __forceinline__ unsigned enc_f32(float v) {
    unsigned u = __float_as_uint(v);
    return (u >> 31) ? ~u : (u | 0x80000000u);
}
__device__ __forceinline__ float dec_f32(unsigned u) {
    return (u & 0x80000000u) ? __uint_as_float(u & 0x7FFFFFFFu)
                             : __uint_as_float(~u);
}
#define ENC_NEG_INF 0x007FFFFFu   // enc(-inf)

// ---------------- K0: reduce al_w/ar_w to per-type vectors, zero counters ----
__global__ void k_init_vecs(const float* __restrict__ al_w,
                            const float* __restrict__ ar_w,
                            float* __restrict__ al_vec,
                            float* __restrict__ ar_vec,
                            int* __restrict__ cnt) {
    int i = blockIdx.x * blockDim.x + threadIdx.x;
    if (i < NTYPES * HID) {                 // al_vec[t][k] = sum_j al_w[t][k][j]
        int t = i / HID, k = i % HID;
        const float* p = al_w + ((size_t)t * HID + k) * HID;
        float s = 0.f;
        for (int j = 0; j < HID; ++j) s += p[j];
        al_vec[i] = s;
    } else if (i < 2 * NTYPES * HID) {
        int j2 = i - NTYPES * HID;
        int t = j2 / HID, k = j2 % HID;
        const float* p = ar_w + ((size_t)t * HID + k) * HID;
        float s = 0.f;
        for (int j = 0; j < HID; ++j) s += p[j];
        ar_vec[j2] = s;
    } else if (i < 2 * NTYPES * HID + NTYPES) {
        cnt[i - 2 * NTYPES * HID] = 0;
    }
}

// ---------------- K1: per-node init of softmax state --------------------------
__global__ void k_init_nodes(float* __restrict__ ssum, unsigned* __restrict__ menc, int nH) {
    int i = blockIdx.x * blockDim.x + threadIdx.x;
    if (i >= nH) return;
    ssum[i] = 0.f;
    menc[i] = ENC_NEG_INF;
}

// ---------------- K2: fill padded sorted-index array with -1 ------------------
__global__ void k_init_sorted(int* __restrict__ sortedIdx, int npad) {
    int i = blockIdx.x * blockDim.x + threadIdx.x;
    if (i < npad) sortedIdx[i] = -1;
}

// ---------------- K3: histogram of node types ---------------------------------
__global__ void k_hist(const int* __restrict__ ntype, int* __restrict__ cnt, int n) {
    int i = blockIdx.x * blockDim.x + threadIdx.x;
    if (i < n) atomicAdd(&cnt[ntype[i]], 1);
}

// ---------------- K4: padded bucket offsets + per-tile type (single thread) ---
__global__ void k_offsets(const int* __restrict__ cnt, int* __restrict__ cursor,
                          int* __restrict__ tileType, int maxTiles) {
    if (blockIdx.x != 0 || threadIdx.x != 0) return;
    int off = 0, tile = 0;
    for (int t = 0; t < NTYPES; ++t) {
        cursor[t] = off;
        int tiles = (cnt[t] + 15) >> 4;
        for (int i = 0; i < tiles && tile < maxTiles; ++i) tileType[tile++] = t;
        off += tiles << 4;
    }
    for (; tile < maxTiles; ++tile) tileType[tile] = -1;
}

// ---------------- K5: scatter node ids into type buckets ----------------------
__global__ void k_scatter(const int* __restrict__ ntype, int* __restrict__ cursor,
                          int* __restrict__ sortedIdx, int n) {
    int i = blockIdx.x * blockDim.x + threadIdx.x;
    if (i >= n) return;
    int pos = atomicAdd(&cursor[ntype[i]], 1);
    sortedIdx[pos] = i;
}

// ---------------- K6: WMMA fp32 GEMM: feat = x@W_t ; accum = x@W_res + b ------
// one block (8 waves) per 16-node tile; wave w owns N columns [16w, 16w+16)
// Padding rows (idx < 0) are clamped to row 0 for A loads: garbage in A row m
// only affects D row m, and those rows are never stored.
__global__ __launch_bounds__(256) void k_gemm_typed(
    const float* __restrict__ x, const float* __restrict__ W_typed,
    const float* __restrict__ W_res, const float* __restrict__ b_res,
    const int* __restrict__ sortedIdx, const int* __restrict__ tileType,
    float* __restrict__ feat, float* __restrict__ accum) {
    int tile = blockIdx.x;
    int ttype = tileType[tile];
    if (ttype < 0) return;

    __shared__ int rows[16];
    int tid = threadIdx.x;
    if (tid < 16) rows[tid] = sortedIdx[tile * 16 + tid];
    __syncthreads();

    int lane  = tid & 31;
    int l15   = lane & 15;
    int khalf = lane >> 4;          // 0: K pair {k0,k0+1}, 1: {k0+2,k0+3}
    int n0    = (tid >> 5) * 16;    // wave id -> 16-wide N slab

    const float* Wt = W_typed + (size_t)ttype * INDIM * OUTDIM;
    int arow = rows[l15];           // A fragment row: m = lane&15
    if (arow < 0) arow = 0;         // clamp: unguarded loads, stores stay masked
    const float* ap = x + (size_t)arow * INDIM + khalf * 2;

    v8f acc_t = {};
    v8f acc_r = {};
    for (int k0 = 0; k0 < INDIM; k0 += 4) {
        v2f a = *(const v2f*)(ap + k0);                 // one b64 load
        int ka = k0 + khalf * 2;
        v2f bt, br;
        bt.x = Wt[(size_t)(ka + 0) * OUTDIM + n0 + l15];
        bt.y = Wt[(size_t)(ka + 1) * OUTDIM + n0 + l15];
        br.x = W_res[(size_t)(ka + 0) * OUTDIM + n0 + l15];
        br.y = W_res[(size_t)(ka + 1) * OUTDIM + n0 + l15];
        if (k0 + 4 < INDIM) {  // pull next B panel lines into cache
            __builtin_prefetch(&Wt[(size_t)(ka + 4) * OUTDIM + n0 + l15], 0, 1);
            __builtin_prefetch(&W_res[(size_t)(ka + 4) * OUTDIM + n0 + l15], 0, 1);
        }
        acc_t = __builtin_amdgcn_wmma_f32_16x16x4_f32(false, a, false, bt,
                                                      (short)0, acc_t, false, false);
        acc_r = __builtin_amdgcn_wmma_f32_16x16x4_f32(false, a, false, br,
                                                      (short)0, acc_r, false, false);
    }

    float bias = b_res[n0 + l15];
    for (int v = 0; v < 8; ++v) {   // C/D layout: m = v + 8*khalf, n = lane&15
        int m = v + 8 * khalf;
        int r = rows[m];
        if (r >= 0) {
            size_t base = (size_t)r * OUTDIM + n0 + l15;
            feat[base]  = acc_t[v];
            accum[base] = acc_r[v] + bias;
        }
    }
}

// ---------------- K7: per-(node,head) attention logits ------------------------
__global__ void k_logits(const float* __restrict__ feat, const int* __restrict__ ntype,
                         const float* __restrict__ al_vec, const float* __restrict__ ar_vec,
                         float* __restrict__ hl, float* __restrict__ hr, int n) {
    int i = blockIdx.x * blockDim.x + threadIdx.x;
    if (i >= n * NHEADS) return;
    int node = i >> 2, head = i & 3;
    int t = ntype[node];
    const float* f  = feat + (size_t)node * OUTDIM + head * HID;
    const float* av = al_vec + t * HID;
    const float* bv = ar_vec + t * HID;
    float sl = 0.f, sr = 0.f;
    for (int d = 0; d < HID; ++d) { float fv = f[d]; sl += fv * av[d]; sr += fv * bv[d]; }
    hl[i] = sl;
    hr[i] = sr;
}

// ---------------- K8: edge logits + segment max -------------------------------
__global__ void k_edge_logit(const int* __restrict__ src, const int* __restrict__ dst,
                             const float* __restrict__ hl, const float* __restrict__ hr,
                             float* __restrict__ ebuf, unsigned* __restrict__ menc, int e) {
    int i = blockIdx.x * blockDim.x + threadIdx.x;
    if (i >= e * NHEADS) return;
    int edge = i >> 2, head = i & 3;
    int s = src[edge], d = dst[edge];
    float v = hl[s * NHEADS + head] + hr[d * NHEADS + head];
    v = (v > 0.f) ? v : NEGSLOPE * v;           // leaky relu
    ebuf[i] = v;
    atomicMax(&menc[d * NHEADS + head], enc_f32(v));
}

// ---------------- K9: exp + segment sum ---------------------------------------
__global__ void k_edge_exp(const int* __restrict__ dst, float* __restrict__ ebuf,
                           const unsigned* __restrict__ menc, float* __restrict__ ssum, int e) {
    int i = blockIdx.x * blockDim.x + threadIdx.x;
    if (i >= e * NHEADS) return;
    int edge = i >> 2, head = i & 3;
    int d = dst[edge];
    float m  = dec_f32(menc[d * NHEADS + head]);
    float ex = __expf(ebuf[i] - m);
    ebuf[i] = ex;
    atomicAdd(&ssum[d * NHEADS + head], ex);
}

// ---------------- K10: message scatter (one wave per edge) --------------------
// accum[dst*128 + d*4 + head] += feat[src*128 + head*32 + d] * alpha[head]
__global__ __launch_bounds__(256) void k_edge_msg(
    const int* __restrict__ src, const int* __restrict__ dst,
    const float* __restrict__ ebuf, const float* __restrict__ ssum,
    const float* __restrict__ feat, float* __restrict__ accum, int e) {
    int widx = (blockIdx.x * blockDim.x + threadIdx.x) >> 5;
    int lane = threadIdx.x & 31;
    if (widx >= e) return;
    int s = src[widx], d = dst[widx];
    float a0 = ebuf[widx * 4 + 0] / ssum[d * NHEADS + 0];
    float a1 = ebuf[widx * 4 + 1] / ssum[d * NHEADS + 1];
    float a2 = ebuf[widx * 4 + 2] / ssum[d * NHEADS + 2];
    float a3 = ebuf[widx * 4 + 3] / ssum[d * NHEADS + 3];
    const float* f = feat + (size_t)s * OUTDIM;
    float* o = accum + (size_t)d * OUTDIM;
    atomicAdd(&o[lane * 4 + 0], f[0 * HID + lane] * a0);
    atomicAdd(&o[lane * 4 + 1], f[1 * HID + lane] * a1);
    atomicAdd(&o[lane * 4 + 2], f[2 * HID + lane] * a2);
    atomicAdd(&o[lane * 4 + 3], f[3 * HID + lane] * a3);
}

// ---------------- K11: final ReLU ---------------------------------------------
__global__ void k_finalize(const float* __restrict__ accum, float* __restrict__ out, int total) {
    int i = blockIdx.x * blockDim.x + threadIdx.x;
    if (i >= total) return;
    float v = accum[i];
    out[i] = v > 0.f ? v : 0.f;
}

static inline size_t align256(size_t x) { return (x + 255) & ~(size_t)255; }

extern "C" void kernel_launch(void* const* d_in, const int* in_sizes, int n_in,
                              void* d_out, int out_size, void* d_ws, size_t ws_size,
                              hipStream_t stream) {
    // input order: x, ntype, etype, src, dst, W_typed, al_w, ar_w, W_res, b_res
    const float* x       = (const float*)d_in[0];
    const int*   ntype   = (const int*)  d_in[1];
    const int*   src     = (const int*)  d_in[3];
    const int*   dst     = (const int*)  d_in[4];
    const float* W_typed = (const float*)d_in[5];
    const float* al_w    = (const float*)d_in[6];
    const float* ar_w    = (const float*)d_in[7];
    const float* W_res   = (const float*)d_in[8];
    const float* b_res   = (const float*)d_in[9];
    float* out = (float*)d_out;

    const int N = in_sizes[1];
    const int E = in_sizes[3];
    const int NPAD = ((N + NTYPES * 15) + 15) & ~15;
    const int maxTiles = NPAD / 16;

    // ---- workspace carve-up ----
    char* p = (char*)d_ws;
    float* feat   = (float*)p;               p += align256((size_t)N * OUTDIM * 4);
    float* accum  = (float*)p;               p += align256((size_t)N * OUTDIM * 4);
    float* hl     = (float*)p;               p += align256((size_t)N * NHEADS * 4);
    float* hr     = (float*)p;               p += align256((size_t)N * NHEADS * 4);
    float* ssum   = (float*)p;               p += align256((size_t)N * NHEADS * 4);
    unsigned* menc = (unsigned*)p;           p += align256((size_t)N * NHEADS * 4);
    float* ebuf   = (float*)p;               p += align256((size_t)E * NHEADS * 4);
    float* al_vec = (float*)p;               p += align256(NTYPES * HID * 4);
    float* ar_vec = (float*)p;               p += align256(NTYPES * HID * 4);
    int* cnt      = (int*)p;                 p += align256(NTYPES * 4);
    int* cursor   = (int*)p;                 p += align256(NTYPES * 4);
    int* tileType = (int*)p;                 p += align256((size_t)maxTiles * 4);
    int* sortedIdx = (int*)p;                p += align256((size_t)NPAD * 4);
    (void)ws_size; (void)n_in; (void)out_size;

    const int B = 256;
    // init
    k_init_vecs<<<(2 * NTYPES * HID + NTYPES + B - 1) / B, B, 0, stream>>>(
        al_w, ar_w, al_vec, ar_vec, cnt);
    k_init_nodes<<<(N * NHEADS + B - 1) / B, B, 0, stream>>>(ssum, menc, N * NHEADS);
    k_init_sorted<<<(NPAD + B - 1) / B, B, 0, stream>>>(sortedIdx, NPAD);
    // type bucketing
    k_hist<<<(N + B - 1) / B, B, 0, stream>>>(ntype, cnt, N);
    k_offsets<<<1, 32, 0, stream>>>(cnt, cursor, tileType, maxTiles);
    k_scatter<<<(N + B - 1) / B, B, 0, stream>>>(ntype, cursor, sortedIdx, N);
    // fp32 WMMA GEMMs (typed projection + residual)
    k_gemm_typed<<<maxTiles, 256, 0, stream>>>(x, W_typed, W_res, b_res,
                                               sortedIdx, tileType, feat, accum);
    // attention logits
    k_logits<<<(N * NHEADS + B - 1) / B, B, 0, stream>>>(feat, ntype, al_vec, ar_vec, hl, hr, N);
    // edge softmax
    k_edge_logit<<<(E * NHEADS + B - 1) / B, B, 0, stream>>>(src, dst, hl, hr, ebuf, menc, E);
    k_edge_exp<<<(E * NHEADS + B - 1) / B, B, 0, stream>>>(dst, ebuf, menc, ssum, E);
    // aggregation (direct transposed layout) + finalize
    k_edge_msg<<<((size_t)E * 32 + B - 1) / B, B, 0, stream>>>(src, dst, ebuf, ssum, feat, accum, E);
    k_finalize<<<(N * OUTDIM + B - 1) / B, B, 0, stream>>>(accum, out, N * OUTDIM);
}